// Tconv_35450660061268
// MI455X (gfx1250) — compile-verified
//
#include <hip/hip_runtime.h>
#include <hip/hip_bf16.h>
#include <stdint.h>

#define H    128
#define KK   4
#define CCH  32      // convtranse channels
#define KS   3
#define NE   40000
#define NR   256
#define TT   4
#define EE   100000
#define NT   1024
#define PIN  (H*KK - H + KK)    // 388
#define CLEN (H - H/KK + 1)     // 97

typedef __attribute__((ext_vector_type(2))) float v2f;
typedef __attribute__((ext_vector_type(8))) float v8f;

__device__ __forceinline__ float sigf(float x) { return 1.f / (1.f + __expf(-x)); }

// ---------------------------------------------------------------------------
// f32 WMMA GEMM, compile-time specialized:
//   C = act( A@op(B) [+ A2@op(B2)] + bias )
// A: MxK row-major.  TB=1: B is NxK row-major (compute A@B^T)
//                    TB=0: B is KxN row-major (compute A@B)
// One wave per 16x16 tile; K-loop unrolled 4x over v_wmma_f32_16x16x4_f32.
// M,N multiples of 16; K,K2 multiples of 16 (true for every call site here).
// act: 0=none 1=relu 2=sigmoid
// ---------------------------------------------------------------------------
template <int TB, bool DUAL, int TB2>
__global__ __launch_bounds__(256) void k_gemm_t(
    const float* __restrict__ A,  const float* __restrict__ B,
    const float* __restrict__ A2, const float* __restrict__ B2,
    const float* __restrict__ bias, float* __restrict__ C,
    int M, int N, int K, int K2, int act)
{
  const int wave = threadIdx.x >> 5;
  const int lane = threadIdx.x & 31;
  const int tilesN = N >> 4;
  const long total = (long)(M >> 4) * tilesN;
  const long tile  = (long)blockIdx.x * 8 + wave;
  if (tile >= total) return;                    // wave-uniform -> EXEC all ones
  const int tm = (int)(tile / tilesN);
  const int tn = (int)(tile % tilesN);
  const int m0 = tm << 4, n0 = tn << 4;
  const int l16  = lane & 15;
  const int half = lane >> 4;
  const int koff = half << 1;                   // lanes 0-15: K{0,1}; lanes 16-31: K{2,3}

  v8f acc = {0.f,0.f,0.f,0.f,0.f,0.f,0.f,0.f};

  {
    const float* ap = A + (size_t)(m0 + l16) * K + koff;
    const float* bp = TB ? (B + (size_t)(n0 + l16) * K + koff)
                         : (B + (size_t)koff * N + (n0 + l16));
    #pragma unroll 4
    for (int k = 0; k < K; k += 4) {
      v2f a, b;
      a.x = ap[k]; a.y = ap[k + 1];
      if (TB) { b.x = bp[k];           b.y = bp[k + 1]; }
      else    { b.x = bp[(size_t)k*N]; b.y = bp[(size_t)(k + 1)*N]; }
      acc = __builtin_amdgcn_wmma_f32_16x16x4_f32(false, a, false, b,
                                                  (short)0, acc, false, false);
    }
  }
  if (DUAL) {
    const float* ap = A2 + (size_t)(m0 + l16) * K2 + koff;
    const float* bp = TB2 ? (B2 + (size_t)(n0 + l16) * K2 + koff)
                          : (B2 + (size_t)koff * N + (n0 + l16));
    #pragma unroll 4
    for (int k = 0; k < K2; k += 4) {
      v2f a, b;
      a.x = ap[k]; a.y = ap[k + 1];
      if (TB2) { b.x = bp[k];           b.y = bp[k + 1]; }
      else     { b.x = bp[(size_t)k*N]; b.y = bp[(size_t)(k + 1)*N]; }
      acc = __builtin_amdgcn_wmma_f32_16x16x4_f32(false, a, false, b,
                                                  (short)0, acc, false, false);
    }
  }

  const int n = n0 + l16;
  const float bv = bias ? bias[n] : 0.f;
  #pragma unroll
  for (int j = 0; j < 8; ++j) {
    int m = m0 + j + (half << 3);               // C/D: VGPR j -> M = m0+j (+8 hi half)
    float v = acc[j] + bv;
    if (act == 1)      v = v > 0.f ? v : 0.f;
    else if (act == 2) v = sigf(v);
    C[(size_t)m * N + n] = v;
  }
}

// Row-wise L2 normalize, H=128. in==out is safe.
__global__ __launch_bounds__(128) void k_norm_rows(const float* __restrict__ in,
                                                   float* __restrict__ out, int rows)
{
  __shared__ float red[4];
  int r = blockIdx.x; if (r >= rows) return;
  int f = threadIdx.x;
  float v = in[(size_t)r * H + f];
  float s = v * v;
  for (int o = 16; o > 0; o >>= 1) s += __shfl_down(s, o, 32);
  if ((threadIdx.x & 31) == 0) red[threadIdx.x >> 5] = s;
  __syncthreads();
  float nrm = sqrtf(red[0] + red[1] + red[2] + red[3]);
  nrm = nrm > 1e-12f ? nrm : 1e-12f;
  out[(size_t)r * H + f] = v / nrm;
}

__global__ void k_copy(const float* __restrict__ a, float* __restrict__ b, int n)
{ int i = blockIdx.x * blockDim.x + threadIdx.x; if (i < n) b[i] = a[i]; }

__global__ void k_zero(float* __restrict__ p, long n)
{ long i = (long)blockIdx.x * blockDim.x + threadIdx.x; if (i < n) p[i] = 0.f; }

// mmt update: two GRUs with h=0, then gated blend. One block of 384 threads.
__global__ __launch_bounds__(384) void k_mmt(
    const float* __restrict__ mmt,
    const float* __restrict__ w1, const float* __restrict__ bi1, const float* __restrict__ bh1,
    const float* __restrict__ w2, const float* __restrict__ bi2, const float* __restrict__ bh2,
    float* __restrict__ out)
{
  __shared__ float x[H];
  __shared__ float g1[3*H], g2[3*H];
  int j = threadIdx.x;
  if (j < H) x[j] = mmt[j];
  __syncthreads();
  float d1 = bi1[j], d2 = bi2[j];
  for (int i = 0; i < H; ++i) { float xi = x[i]; d1 += xi * w1[j*H+i]; d2 += xi * w2[j*H+i]; }
  g1[j] = d1; g2[j] = d2;
  __syncthreads();
  if (j < H) {
    float r1 = sigf(g1[j] + bh1[j]);
    float z1 = sigf(g1[H+j] + bh1[H+j]);
    float n1 = tanhf(g1[2*H+j] + r1 * bh1[2*H+j]);
    float nm = (1.f - z1) * n1;                       // h==0
    float r2 = sigf(g2[j] + bh2[j]);
    float z2 = sigf(g2[H+j] + bh2[H+j]);
    float n2 = tanhf(g2[2*H+j] + r2 * bh2[2*H+j]);
    float gg = (1.f - z2) * n2;
    float m = x[j];
    out[j] = m + gg * (nm - m);
  }
}

// v = mmt @ tc_w^T + tc_b   (the KK x (H/KK) conv filters, flattened)
__global__ __launch_bounds__(128) void k_tconv_filters(
    const float* __restrict__ mmt, const float* __restrict__ w,
    const float* __restrict__ b, float* __restrict__ v)
{
  __shared__ float x[H];
  int j = threadIdx.x;
  x[j] = mmt[j];
  __syncthreads();
  float acc = b[j];
  for (int i = 0; i < H; ++i) acc += x[i] * w[j*H+i];
  v[j] = acc;
}

// Fold conv(VALID)+pool into one 128x128 matrix:
// W2[o,m] = sum_{k,i, 0<=m-i<97} v[k,i] * pw[o, k*97 + (m-i)]
__global__ __launch_bounds__(128) void k_build_w2(
    const float* __restrict__ v, const float* __restrict__ pw, float* __restrict__ w2)
{
  __shared__ float vs[H];
  int o = blockIdx.x, m = threadIdx.x;
  vs[m] = v[m];
  __syncthreads();
  float acc = 0.f;
  for (int k = 0; k < KK; ++k)
    for (int i = 0; i < H/KK; ++i) {
      int jj = m - i;
      if (jj >= 0 && jj < CLEN) acc += vs[k*(H/KK)+i] * pw[(size_t)o*PIN + k*CLEN + jj];
    }
  w2[o*H + m] = acc;
}

__global__ void k_scatter_mask(uint8_t* __restrict__ mask,
                               const int* __restrict__ s, const int* __restrict__ d,
                               const int* __restrict__ r)
{
  int e = blockIdx.x * blockDim.x + threadIdx.x;
  if (e < EE) {
    size_t base = (size_t)r[e] * NE;
    mask[base + s[e]] = 1;
    mask[base + d[e]] = 1;
  }
}

// Per-relation mean of touched entity rows (deduped via mask), LDS-compacted scan.
__global__ __launch_bounds__(128) void k_agg_rel(
    const uint8_t* __restrict__ mask, const float* __restrict__ ent,
    float* __restrict__ rel_ent)
{
  __shared__ int list[128];
  __shared__ int cnt_s;
  int r = blockIdx.x, f = threadIdx.x;
  const uint8_t* mr = mask + (size_t)r * NE;
  float acc = 0.f; int total = 0;
  for (int base = 0; base < NE; base += 128) {
    if (f == 0) cnt_s = 0;
    __syncthreads();
    int nn = base + f;
    if (nn < NE && mr[nn]) { int p = atomicAdd(&cnt_s, 1); list[p] = nn; }
    __syncthreads();
    int c = cnt_s;
    for (int i = 0; i < c; ++i) acc += ent[(size_t)list[i] * H + f];
    total += c;
    __syncthreads();
  }
  rel_ent[(size_t)r * H + f] = (total > 0) ? acc / (float)total : 0.f;
}

__global__ __launch_bounds__(256) void k_concat(const float* __restrict__ a,
                                                const float* __restrict__ b,
                                                float* __restrict__ out)
{
  int r = blockIdx.x, j = threadIdx.x;
  out[(size_t)r * (2*H) + j] = (j < H) ? a[(size_t)r*H + j] : b[(size_t)r*H + (j - H)];
}

__global__ __launch_bounds__(128) void k_gru_gate(const float* __restrict__ gi,
                                                  const float* __restrict__ gh,
                                                  const float* __restrict__ hprev,
                                                  float* __restrict__ out)
{
  int r = blockIdx.x, j = threadIdx.x;
  const float* gir = gi + (size_t)r * 3*H;
  const float* ghr = gh + (size_t)r * 3*H;
  float rr = sigf(gir[j]       + ghr[j]);
  float z  = sigf(gir[H + j]   + ghr[H + j]);
  float n  = tanhf(gir[2*H + j] + rr * ghr[2*H + j]);
  float h  = hprev[(size_t)r*H + j];
  out[(size_t)r*H + j] = (1.f - z) * n + z * h;
}

// RGCN message aggregation: sums[d] += h[s] + nrel[rid]; cnt[d] += 1
__global__ __launch_bounds__(128) void k_edge_scatter(
    const float* __restrict__ h, const float* __restrict__ ef,
    const int* __restrict__ s, const int* __restrict__ d, const int* __restrict__ r,
    float* __restrict__ sums, float* __restrict__ cnt)
{
  int e = blockIdx.x, f = threadIdx.x;
  int de = d[e];
  float v = h[(size_t)s[e] * H + f] + ef[(size_t)r[e] * H + f];
  atomicAdd(&sums[(size_t)de * H + f], v);
  if (f == 0) atomicAdd(&cnt[de], 1.f);
}

__global__ void k_divide(float* __restrict__ sums, const float* __restrict__ cnt, long n)
{
  long i = (long)blockIdx.x * blockDim.x + threadIdx.x;
  if (i < n) sums[i] = sums[i] / fmaxf(cnt[i >> 7], 1.f);
}

__global__ void k_gate_ent(float* __restrict__ ent, const float* __restrict__ u,
                           const float* __restrict__ nf, long n)
{
  long i = (long)blockIdx.x * blockDim.x + threadIdx.x;
  if (i < n) { float e = ent[i]; ent[i] = e + u[i] * (nf[i] - e); }
}

// convtranse front: gather two rows, pad-1 conv over channels, relu
__global__ __launch_bounds__(128) void k_convtr(
    const float* __restrict__ tabA, const float* __restrict__ tabB,
    const int* __restrict__ idxA, const int* __restrict__ idxB,
    const float* __restrict__ ck, const float* __restrict__ cb,
    float* __restrict__ y)
{
  int n = blockIdx.x, c = blockIdx.y, p = threadIdx.x;
  const float* r0 = tabA + (size_t)idxA[n] * H;
  const float* r1 = tabB + (size_t)idxB[n] * H;
  float acc = cb[c];
  #pragma unroll
  for (int s = 0; s < KS; ++s) {
    int q = p + s - 1;
    if (q >= 0 && q < H)
      acc += r0[q] * ck[c*2*KS + s] + r1[q] * ck[c*2*KS + KS + s];
  }
  y[((size_t)n * CCH + c) * H + p] = fmaxf(acc, 0.f);
}

// ---------------------------------------------------------------------------
extern "C" void kernel_launch(void* const* d_in, const int* in_sizes, int n_in,
                              void* d_out, int out_size, void* d_ws, size_t ws_size,
                              hipStream_t stream) {
  (void)in_sizes; (void)n_in; (void)out_size; (void)ws_size;
  const float* ent_embeds = (const float*)d_in[0];
  const float* rel_embeds = (const float*)d_in[1];
  const float* mmt_embed  = (const float*)d_in[2];
  const float* tc_w  = (const float*)d_in[3];
  const float* tc_b  = (const float*)d_in[4];
  const float* tc_pw = (const float*)d_in[5];
  const float* tc_pb = (const float*)d_in[6];
  const float* g1_wih = (const float*)d_in[7];
  const float* g1_whh = (const float*)d_in[8];
  const float* g1_bih = (const float*)d_in[9];
  const float* g1_bhh = (const float*)d_in[10];
  const float* l1_wih = (const float*)d_in[11];
  const float* l1_bih = (const float*)d_in[13];
  const float* l1_bhh = (const float*)d_in[14];
  const float* l2_wih = (const float*)d_in[15];
  const float* l2_bih = (const float*)d_in[17];
  const float* l2_bhh = (const float*)d_in[18];
  const float* lin_w = (const float*)d_in[19];
  const float* lin_b = (const float*)d_in[20];
  const float* rgcn_wn = (const float*)d_in[21];
  const float* rgcn_ws = (const float*)d_in[22];
  const float* rgcn_b  = (const float*)d_in[23];
  const float* oc_k  = (const float*)d_in[24];
  const float* oc_kb = (const float*)d_in[25];
  const float* oc_fw = (const float*)d_in[26];
  const float* oc_fb = (const float*)d_in[27];
  const float* rc_k  = (const float*)d_in[28];
  const float* rc_kb = (const float*)d_in[29];
  const float* rc_fw = (const float*)d_in[30];
  const float* rc_fb = (const float*)d_in[31];
  const int* src  = (const int*)d_in[32];
  const int* dst  = (const int*)d_in[33];
  const int* rid  = (const int*)d_in[34];
  const int* subj = (const int*)d_in[35];
  const int* rel  = (const int*)d_in[36];
  const int* obj  = (const int*)d_in[37];

  float* ws = (float*)d_ws;
  size_t off = 0;
  auto alloc = [&](size_t n) { float* p = ws + off; off += n; return p; };
  float* E0   = alloc((size_t)NE*H);
  float* E1   = alloc((size_t)NE*H);
  float* WK1  = alloc((size_t)NE*H);
  float* WK2  = alloc((size_t)NE*H);
  float* SUM  = alloc((size_t)NE*H);   // overlaid: edge sums / rel-mask / gate U / decoder conv buf
  float* CNT  = alloc(NE);
  float* R0   = alloc((size_t)NR*H);
  float* R1   = alloc((size_t)NR*H);
  float* MMT  = alloc(H);
  float* MMT2 = alloc(H);
  float* V    = alloc(H);
  float* W2   = alloc(H*H);
  float* RELE = alloc((size_t)NR*H);
  float* GIN  = alloc((size_t)NR*2*H);
  float* GI   = alloc((size_t)NR*3*H);
  float* GH   = alloc((size_t)NR*3*H);
  float* QO   = alloc((size_t)NT*H);
  float* QR   = alloc((size_t)NT*H);

  // A @ B^T (+bias, act): B is (N,K) row-major weights
  auto gemm_bt = [&](const float* A, const float* B, const float* bias, float* C,
                     int M, int N, int K, int act) {
    long tiles = (long)(M/16) * (N/16);
    int blocks = (int)((tiles + 7) / 8);
    k_gemm_t<1, false, 0><<<blocks, 256, 0, stream>>>(
        A, B, nullptr, nullptr, bias, C, M, N, K, 0, act);
  };
  // A @ B + A2 @ B2 (+bias, act): B, B2 are (K,N) row-major
  auto gemm_dual = [&](const float* A, const float* B, const float* A2, const float* B2,
                       const float* bias, float* C, int M, int N, int K, int K2, int act) {
    long tiles = (long)(M/16) * (N/16);
    int blocks = (int)((tiles + 7) / 8);
    k_gemm_t<0, true, 0><<<blocks, 256, 0, stream>>>(
        A, B, A2, B2, bias, C, M, N, K, K2, act);
  };

  // init: normalized tables, mmt copy
  k_norm_rows<<<NE, 128, 0, stream>>>(ent_embeds, E0, NE);
  k_norm_rows<<<NR, 128, 0, stream>>>(rel_embeds, R0, NR);
  k_copy<<<1, 128, 0, stream>>>(mmt_embed, MMT, H);

  float* entP = E0;  float* entQ = E1;
  float* relP = R0;  float* relQ = R1;
  float* mmtP = MMT; float* mmtQ = MMT2;

  for (int t = 0; t < TT; ++t) {
    const int* st = src + (size_t)t * EE;
    const int* dt = dst + (size_t)t * EE;
    const int* rt = rid + (size_t)t * EE;

    // mmt <- gated GRU blend
    k_mmt<<<1, 384, 0, stream>>>(mmtP, l1_wih, l1_bih, l1_bhh, l2_wih, l2_bih, l2_bhh, mmtQ);
    { float* tmp = mmtP; mmtP = mmtQ; mmtQ = tmp; }

    // tconv folded into a single 128x128 matrix, then WMMA GEMM
    k_tconv_filters<<<1, 128, 0, stream>>>(mmtP, tc_w, tc_b, V);
    k_build_w2<<<H, 128, 0, stream>>>(V, tc_pw, W2);
    gemm_bt(entP, W2, tc_pb, entQ, NE, H, H, 0);
    { float* tmp = entP; entP = entQ; entQ = tmp; }
    gemm_bt(relP, W2, tc_pb, relQ, NR, H, H, 0);
    { float* tmp = relP; relP = relQ; relQ = tmp; }

    // relation aggregation (deduped mean over touched nodes)
    uint8_t* MASK = (uint8_t*)SUM;
    k_zero<<<(int)(((long)NR*NE/4 + 255)/256), 256, 0, stream>>>((float*)MASK, (long)NR*NE/4);
    k_scatter_mask<<<(EE + 255)/256, 256, 0, stream>>>(MASK, st, dt, rt);
    k_agg_rel<<<NR, 128, 0, stream>>>(MASK, entP, RELE);

    // relation GRU (WMMA GEMMs for both projections) + normalize
    k_concat<<<NR, 256, 0, stream>>>(RELE, rel_embeds, GIN);
    gemm_bt(GIN,  g1_wih, g1_bih, GI, NR, 3*H, 2*H, 0);
    gemm_bt(relP, g1_whh, g1_bhh, GH, NR, 3*H, H,   0);
    k_gru_gate<<<NR, 128, 0, stream>>>(GI, GH, relP, relQ);
    k_norm_rows<<<NR, 128, 0, stream>>>(relQ, relQ, NR);
    { float* tmp = relP; relP = relQ; relQ = tmp; }   // relP = n_rel

    // RGCN: 2 layers, each = atomics-aggregate + fused dual WMMA GEMM with relu
    const float* hcur = entP;
    float* houts[2] = { WK1, WK2 };
    for (int l = 0; l < 2; ++l) {
      k_zero<<<(int)(((long)NE*H + 255)/256), 256, 0, stream>>>(SUM, (long)NE*H);
      k_zero<<<(NE + 255)/256, 256, 0, stream>>>(CNT, NE);
      k_edge_scatter<<<EE, 128, 0, stream>>>(hcur, relP, st, dt, rt, SUM, CNT);
      k_divide<<<(int)(((long)NE*H + 255)/256), 256, 0, stream>>>(SUM, CNT, (long)NE*H);
      gemm_dual(SUM, rgcn_wn + (size_t)l*H*H, hcur, rgcn_ws + (size_t)l*H*H,
                rgcn_b + (size_t)l*H, houts[l], NE, H, H, H, 1);
      hcur = houts[l];
    }
    k_norm_rows<<<NE, 128, 0, stream>>>(WK2, WK2, NE);             // nf
    gemm_bt(entP, lin_w, lin_b, SUM, NE, H, H, 2);                 // U = sigmoid(ent@lin^T+b)
    k_gate_ent<<<(int)(((long)NE*H + 255)/256), 256, 0, stream>>>(entP, SUM, WK2, (long)NE*H);
  }

  // decoders
  float* obj_logit = (float*)d_out;
  float* rel_logit = obj_logit + (size_t)NT * NE;
  float* CONV = SUM;   // NT*CCH*H = 4.19M floats <= NE*H

  k_convtr<<<dim3(NT, CCH), 128, 0, stream>>>(entP, relP, subj, rel, oc_k, oc_kb, CONV);
  gemm_bt(CONV, oc_fw, oc_fb, QO, NT, H, CCH*H, 1);
  gemm_bt(QO, entP, nullptr, obj_logit, NT, NE, H, 0);

  k_convtr<<<dim3(NT, CCH), 128, 0, stream>>>(entP, entP, subj, obj, rc_k, rc_kb, CONV);
  gemm_bt(CONV, rc_fw, rc_fb, QR, NT, H, CCH*H, 1);
  gemm_bt(QR, relP, nullptr, rel_logit, NT, NR, H, 0);
}